// UniformAffineQuantizer_23098334118321
// MI455X (gfx1250) — compile-verified
//
#include <hip/hip_runtime.h>
#include <math.h>

#define GROUP 128
#define QMAXF 15.0f
#define CLIPMIN 1e-4f
#define CLIPMAX 1e4f

// ds_swizzle_b32 in group-of-32 XOR mode: offset = {1'b0, xor[4:0], or[4:0]=0, and[4:0]=0x1f}
template <int MASK>
__device__ __forceinline__ float swz_xor(float v) {
    int r = __builtin_amdgcn_ds_swizzle(__float_as_int(v), (MASK << 10) | 0x1F);
    return __int_as_float(r);
}

__device__ __forceinline__ float clampf(float v, float lo, float hi) {
    return __builtin_amdgcn_fmed3f(v, lo, hi);   // single v_med3_f32
}

// One wave32 per 128-element group: lane l owns float4 at group_base + 4*l.
// Wave-wide min/max via 5-step ds_swizzle XOR butterfly (single DS op per step).
__global__ __launch_bounds__(256) void
fakequant_g128_kernel(const float* __restrict__ x,
                      const float* __restrict__ up,
                      const float* __restrict__ low,
                      float* __restrict__ out,
                      int n_groups)
{
    const int lane = threadIdx.x & 31;
    const int wid  = threadIdx.x >> 5;
    const int wpb  = blockDim.x >> 5;               // waves per block (8)
    const int stride = (int)gridDim.x * wpb;        // total waves in grid

    for (int g = (int)blockIdx.x * wpb + wid; g < n_groups; g += stride) {
        const int idx = (g << 5) + lane;            // float4 index, fits in 32 bits

        // 16B-aligned 128-bit load: wave reads 512 contiguous bytes.
        float4 v = reinterpret_cast<const float4*>(x)[idx];

        // Prefetch next grid-stride iteration's line (global_prefetch_b8).
        if (g + stride < n_groups) {
            __builtin_prefetch(reinterpret_cast<const float4*>(x) + idx + (stride << 5), 0, 0);
        }

        // In-lane reduction over 4 elements.
        float vmin = fminf(fminf(v.x, v.y), fminf(v.z, v.w));
        float vmax = fmaxf(fmaxf(v.x, v.y), fmaxf(v.z, v.w));

        // Cross-lane XOR butterfly over the 32 lanes of the wave.
        vmin = fminf(vmin, swz_xor<16>(vmin));
        vmax = fmaxf(vmax, swz_xor<16>(vmax));
        vmin = fminf(vmin, swz_xor<8>(vmin));
        vmax = fmaxf(vmax, swz_xor<8>(vmax));
        vmin = fminf(vmin, swz_xor<4>(vmin));
        vmax = fmaxf(vmax, swz_xor<4>(vmax));
        vmin = fminf(vmin, swz_xor<2>(vmin));
        vmax = fmaxf(vmax, swz_xor<2>(vmax));
        vmin = fminf(vmin, swz_xor<1>(vmin));
        vmax = fmaxf(vmax, swz_xor<1>(vmax));

        // Per-group learned bound factors (uniform across the wave; broadcast load).
        const float uf = up[g];
        const float lf = low[g];
        const float sig_up  = 1.0f / (1.0f + __expf(-uf));
        const float sig_low = 1.0f / (1.0f + __expf(-lf));

        const float xmax = sig_up  * vmax;
        const float xmin = sig_low * vmin;

        const float scale     = clampf((xmax - xmin) * (1.0f / QMAXF), CLIPMIN, CLIPMAX);
        const float inv_scale = 1.0f / scale;       // single division per group
        const float round_zp  = rintf(clampf(-xmin * inv_scale, -CLIPMAX, CLIPMAX));

        // Fake-quant each of the lane's 4 elements (rintf == round-half-even,
        // clamp == v_med3_f32): 6 VALU ops per element.
        float4 r;
        float t;
        t   = clampf(rintf(v.x * inv_scale) + round_zp, 0.0f, QMAXF);
        r.x = (t - round_zp) * scale;
        t   = clampf(rintf(v.y * inv_scale) + round_zp, 0.0f, QMAXF);
        r.y = (t - round_zp) * scale;
        t   = clampf(rintf(v.z * inv_scale) + round_zp, 0.0f, QMAXF);
        r.z = (t - round_zp) * scale;
        t   = clampf(rintf(v.w * inv_scale) + round_zp, 0.0f, QMAXF);
        r.w = (t - round_zp) * scale;

        // 128-bit coalesced store.
        reinterpret_cast<float4*>(out)[idx] = r;
    }
}

extern "C" void kernel_launch(void* const* d_in, const int* in_sizes, int n_in,
                              void* d_out, int out_size, void* d_ws, size_t ws_size,
                              hipStream_t stream)
{
    (void)n_in; (void)out_size; (void)d_ws; (void)ws_size;

    const float* x   = (const float*)d_in[0];
    const float* up  = (const float*)d_in[1];   // n_groups elements
    const float* low = (const float*)d_in[2];   // n_groups elements
    float* out = (float*)d_out;

    const int n_groups = in_sizes[1];           // = dim0 * dim1 / 128

    const int threads = 256;                    // 8 wave32 per block
    const int wpb = threads / 32;
    int blocks = (n_groups + wpb - 1) / wpb;
    if (blocks > 8192) blocks = 8192;           // grid-stride beyond this
    if (blocks < 1) blocks = 1;

    fakequant_g128_kernel<<<blocks, threads, 0, stream>>>(x, up, low, out, n_groups);
}